// YOLOv8Loss_40939628265908
// MI455X (gfx1250) — compile-verified
//
#include <hip/hip_runtime.h>

// YOLO loss on MI455X (gfx1250, wave32).
// Bandwidth-bound streaming reduction: 90 MB in -> 1 float out.
// WMMA f32_16x16x4 (all-ones B) used as a full-precision cross-lane f32 reducer.

typedef __attribute__((ext_vector_type(2))) float v2f;
typedef __attribute__((ext_vector_type(8))) float v8f;

#define NCELLS (16384 * 7 * 7)   // 802,816 cells, 14 f32 channels each
#define NBLK   1024
#define NTHR   256
#define NWAVES (NTHR / 32)

// Reduce one f32 value per lane across a full wave32 using
// V_WMMA_F32_16X16X4_F32: D = A(16x4) x B(4x16, all ones) + 0.
// A layout (32-bit A, 16x4): lane L<16 holds A[L, K=0..1] in its 2 VGPRs,
// lane L>=16 holds A[L-16, K=2..3]. We put the accumulator in the first of
// the pair and zero in the second, so rowsum_m = acc_m + acc_{m+16}.
// D layout: lane L<16, VGPR j = D[j, L]; lane L>=16, VGPR j = D[j+8, L-16].
// With all-ones B every column of D equals the rowsum vector, so
// sum(own 8 regs) gives rows 0..7 (lanes 0-15) or rows 8..15 (lanes 16-31);
// one shfl_xor(16) completes the 32-lane total on every lane.
// Caller must guarantee the wave is fully active (EXEC == all ones).
__device__ __forceinline__ float wave_reduce_wmma(float v) {
    v2f a; a[0] = v;    a[1] = 0.0f;
    v2f b; b[0] = 1.0f; b[1] = 1.0f;
    v8f c = {};
    v8f d = __builtin_amdgcn_wmma_f32_16x16x4_f32(
        /*neg_a=*/false, a, /*neg_b=*/false, b,
        /*c_mod=*/(short)0, c, /*reuse_a=*/false, /*reuse_b=*/false);
    float s = ((d[0] + d[1]) + (d[2] + d[3])) + ((d[4] + d[5]) + (d[6] + d[7]));
    s += __shfl_xor(s, 16, 32);
    return s;
}

__device__ __forceinline__ float cell_loss(const float* __restrict__ pred,
                                           const float* __restrict__ targ,
                                           int cid) {
    // 14 channels per cell, base byte offset = 56*cid -> float2 aligned.
    const float2* P = reinterpret_cast<const float2*>(pred) + cid * 7;
    const float2* T = reinterpret_cast<const float2*>(targ) + cid * 7;
    float2 p01 = P[0], p23 = P[1], p45 = P[2], p67 = P[3],
           p89 = P[4], pab = P[5], pcd = P[6];
    float2 t01 = T[0], t23 = T[1], t45 = T[2], t67 = T[3],
           t89 = T[4], tab = T[5], tcd = T[6];

    const float p0 = p01.x, p1 = p01.y, p2 = p23.x, p3 = p23.y, p4 = p45.x;
    const float p5 = p45.y, p6 = p67.x, p7 = p67.y, p8 = p89.x, p9 = p89.y;
    const float p10 = pab.x, p11 = pab.y, p12 = pcd.x, p13 = pcd.y;
    const float t0 = t01.x, t1 = t01.y, t2 = t23.x, t3 = t23.y, t4 = t45.x;
    const float t5 = t45.y, t6 = t67.x, t7 = t67.y, t8 = t89.x, t9 = t89.y;
    const float t10 = tab.x, t11 = tab.y, t12 = tcd.x, t13 = tcd.y;

    const float S = 7.0f;
    const float conf_t = t4;
    const float coord = (conf_t > 0.0f) ? 1.0f : 0.0f;

    // Target box 0 corners / area (computed exactly as the reference does).
    const float txy_x = t0 / S, txy_y = t1 / S;
    const float tlt_x = txy_x - 0.5f * t2, tlt_y = txy_y - 0.5f * t3;
    const float trb_x = txy_x + 0.5f * t2, trb_y = txy_y + 0.5f * t3;
    const float area_t = (trb_x - tlt_x) * (trb_y - tlt_y);

    // Predicted box 0
    const float q0x = p0 / S, q0y = p1 / S;
    const float lt0x = q0x - 0.5f * p2, lt0y = q0y - 0.5f * p3;
    const float rb0x = q0x + 0.5f * p2, rb0y = q0y + 0.5f * p3;
    const float w0 = fmaxf(fminf(rb0x, trb_x) - fmaxf(lt0x, tlt_x), 0.0f);
    const float h0 = fmaxf(fminf(rb0y, trb_y) - fmaxf(lt0y, tlt_y), 0.0f);
    const float inter0 = w0 * h0;
    const float ap0 = (rb0x - lt0x) * (rb0y - lt0y);
    const float iou0 = inter0 / (ap0 + area_t - inter0);

    // Predicted box 1
    const float q1x = p5 / S, q1y = p6 / S;
    const float lt1x = q1x - 0.5f * p7, lt1y = q1y - 0.5f * p8;
    const float rb1x = q1x + 0.5f * p7, rb1y = q1y + 0.5f * p8;
    const float w1 = fmaxf(fminf(rb1x, trb_x) - fmaxf(lt1x, tlt_x), 0.0f);
    const float h1 = fmaxf(fminf(rb1y, trb_y) - fmaxf(lt1y, tlt_y), 0.0f);
    const float inter1 = w1 * h1;
    const float ap1 = (rb1x - lt1x) * (rb1y - lt1y);
    const float iou1 = inter1 / (ap1 + area_t - inter1);

    // argmax (first-max tie-break, like jnp.argmax) + responsible box select
    const bool sel1 = iou1 > iou0;
    const float max_iou = sel1 ? iou1 : iou0;
    const float rp0 = sel1 ? p5 : p0, rp1 = sel1 ? p6 : p1, rp2 = sel1 ? p7 : p2;
    const float rp3 = sel1 ? p8 : p3, rp4 = sel1 ? p9 : p4;
    const float rt0 = sel1 ? t5 : t0, rt1 = sel1 ? t6 : t1, rt2 = sel1 ? t7 : t2;
    const float rt3 = sel1 ? t8 : t3, rt4 = sel1 ? t9 : t4;
    (void)rt4;

    const float dx = rp0 - rt0, dy = rp1 - rt1;
    const float lxy = dx * dx + dy * dy;
    const float sw = sqrtf(rp2) - sqrtf(rt2);
    const float sh = sqrtf(rp3) - sqrtf(rt3);
    const float lwh = sw * sw + sh * sh;
    const float dob = rp4 - max_iou;
    const float lobj = dob * dob;

    // BCE over 4 class channels (10..13)
    float lcls = 0.0f;
    lcls -= t10 * logf(p10) + (1.0f - t10) * logf(1.0f - p10);
    lcls -= t11 * logf(p11) + (1.0f - t11) * logf(1.0f - p11);
    lcls -= t12 * logf(p12) + (1.0f - t12) * logf(1.0f - p12);
    lcls -= t13 * logf(p13) + (1.0f - t13) * logf(1.0f - p13);

    // focal-df term on confidence channel 4 (applies to ALL cells, unmasked)
    const float qv = conf_t, pv = p4;
    const float alpha = (1.0f - qv) / (1.0f - pv);
    const float ldf = alpha * (pv - qv) * logf(pv) + (qv - pv) * logf(1.0f - pv);

    return coord * (lxy + lwh + lobj + lcls) + ldf;
}

__global__ void __launch_bounds__(NTHR)
yolo_loss_partial(const float* __restrict__ pred, const float* __restrict__ targ,
                  float* __restrict__ partials) {
    const int tid = blockIdx.x * NTHR + threadIdx.x;
    const int stride = gridDim.x * NTHR;
    float acc = 0.0f;
    for (int i = tid; i < NCELLS; i += stride)
        acc += cell_loss(pred, targ, i);

    // wave-level f32 reduction via WMMA (wave fully active here)
    const float ws = wave_reduce_wmma(acc);

    __shared__ float lds[NWAVES];
    const int lane = threadIdx.x & 31;
    const int wid = threadIdx.x >> 5;
    if (lane == 0) lds[wid] = ws;
    __syncthreads();
    if (wid == 0) {  // whole wave 0 enters: EXEC all ones for the WMMA below
        const float v = (lane < NWAVES) ? lds[lane] : 0.0f;
        const float bs = wave_reduce_wmma(v);
        if (lane == 0) partials[blockIdx.x] = bs;
    }
}

__global__ void __launch_bounds__(NTHR)
yolo_loss_final(const float* __restrict__ partials, float* __restrict__ out) {
    float acc = 0.0f;
    for (int i = threadIdx.x; i < NBLK; i += NTHR)
        acc += partials[i];

    const float ws = wave_reduce_wmma(acc);

    __shared__ float lds[NWAVES];
    const int lane = threadIdx.x & 31;
    const int wid = threadIdx.x >> 5;
    if (lane == 0) lds[wid] = ws;
    __syncthreads();
    if (wid == 0) {
        const float v = (lane < NWAVES) ? lds[lane] : 0.0f;
        const float bs = wave_reduce_wmma(v);
        if (lane == 0) out[0] = bs;
    }
}

extern "C" void kernel_launch(void* const* d_in, const int* in_sizes, int n_in,
                              void* d_out, int out_size, void* d_ws, size_t ws_size,
                              hipStream_t stream) {
    const float* pred = (const float*)d_in[0];
    const float* targ = (const float*)d_in[1];
    float* out = (float*)d_out;
    float* partials = (float*)d_ws;  // NBLK floats = 4 KB, fully rewritten every call

    yolo_loss_partial<<<NBLK, NTHR, 0, stream>>>(pred, targ, partials);
    yolo_loss_final<<<1, NTHR, 0, stream>>>(partials, out);
}